// HH_Synaptic_9045201126062
// MI455X (gfx1250) — compile-verified
//
#include <hip/hip_runtime.h>
#include <cstdint>
#include <cstddef>

// Problem constants (match reference)
#define B_    16
#define N_    2000
#define L_    1024
#define BLOCK 128
#define DEPTH 4      // async pipeline depth (buffers / outstanding loads)

// ---- fast math primitives: single-instruction trans ops -------------------
__device__ __forceinline__ float fexp(float x) {
  // exp(x) = exp2(x * log2(e)) -> v_exp_f32
  return __builtin_amdgcn_exp2f(x * 1.44269504088896340736f);
}
__device__ __forceinline__ float frcp(float x) {
  // v_rcp_f32 (fast approx; avoids ~10-instruction IEEE div expansion)
  return __builtin_amdgcn_rcpf(x);
}

// Crank-Nicolson gate update: x' = (a*DT + (1 - DT/2*(a+b))*x) / (1 + DT/2*(a+b))
__device__ __forceinline__ float gate_update(float a, float b, float x) {
  float s = 0.01f * (a + b);                       // DT/2 = 0.01
  return __builtin_fmaf(0.02f, a, (1.0f - s) * x) * frcp(1.0f + s);
}

__device__ __forceinline__ float vsig(float V) {
  // sigmoid((V - VT)/KP), VT=-20, KP=3
  return frcp(1.0f + fexp((-20.0f - V) * (1.0f / 3.0f)));
}

__global__ __launch_bounds__(BLOCK) void hh_synaptic_kernel(
    const float* __restrict__ z, float* __restrict__ out) {
  // Per-wave-private ring of DEPTH buffers for async z streaming. No
  // cross-wave sharing -> no barriers; sync is s_wait_asynccnt/dscnt only.
  __shared__ float zbuf[DEPTH][BLOCK];

  const int tid = threadIdx.x;
  const int gid = blockIdx.x * BLOCK + tid;        // 0 .. B*L-1
  const int b   = gid >> 10;                       // / L (L = 1024)
  const int l   = gid & (L_ - 1);
  const size_t base = (size_t)b * (size_t)N_ * (size_t)L_ + (size_t)l;

  // This lane's slot-0 LDS byte offset (generic-cast truncation = LDS offset).
  const uint32_t lds_lane = (uint32_t)(uintptr_t)(&zbuf[0][0]) + (uint32_t)tid * 4u;

  // Warm the pipeline: issue async loads of rows 0..DEPTH-1.
#pragma unroll
  for (int r = 0; r < DEPTH; ++r) {
    asm volatile("global_load_async_to_lds_b32 %0, %1, off th:TH_LOAD_NT"
                 :: "v"(lds_lane + (uint32_t)r * (BLOCK * 4u)),
                    "v"(z + base + (size_t)r * (size_t)L_)
                 : "memory");
  }

  // Initial carry
  float V = -70.0f, m = 0.0f, n = 0.0f, h = 1.0f, y = 0.0f;

  // t = 0 output row (V0 = -70 everywhere)
  __builtin_nontemporal_store(vsig(V), &out[base]);

  for (int t = 1; t < N_; ++t) {
    // Consume slot (t-1)%DEPTH. Note (t-1+DEPTH)%DEPTH == (t-1)%DEPTH, so the
    // refill prefetch reuses the exact same LDS offset after the ds_load.
    const uint32_t off = lds_lane + (uint32_t)(((t - 1) & (DEPTH - 1)) * (BLOCK * 4));

    float zprev;
    asm volatile("s_wait_asynccnt 0x3\n\t"   // oldest of DEPTH outstanding done
                 "ds_load_b32 %0, %1\n\t"
                 "s_wait_dscnt 0x0"
                 : "=v"(zprev) : "v"(off) : "memory");

    // Refill: prefetch row t+DEPTH-1 (clamped in-bounds; tail rows redundant
    // but harmless) into the slot we just freed.
    int pr = t + (DEPTH - 1);
    pr = pr < N_ ? pr : (N_ - 1);
    asm volatile("global_load_async_to_lds_b32 %0, %1, off th:TH_LOAD_NT"
                 :: "v"(off), "v"(z + base + (size_t)pr * (size_t)L_)
                 : "memory");

    // ---- membrane update -------------------------------------------------
    float m3 = m * m * m;
    float p1 = 40.0f * m3 * h;                     // GNA * m^3 * h
    float n2 = n * n;
    float p2 = 35.0f * n2 * n2;                    // GK * n^4
    float G  = 0.01f * (p1 + p2 + 0.3f + y);       // DT/2*(p1+p2+GL+GS*y)
    float E  = __builtin_fmaf(p1, 55.0f, __builtin_fmaf(p2, -77.0f, -19.5f));
    float Vn = __builtin_fmaf(V, 1.0f - G, 0.02f * (E + 1.0f)) * frcp(1.0f + G);

    // ---- rate constants (singularity-safe selects) -----------------------
    const bool at25  = (Vn == 25.0f);
    const bool atm35 = (Vn == -35.0f);
    float e1  = fexp((25.0f - Vn)  * (1.0f / 9.0f));   // exp((-V+25)/9)
    float e2  = fexp((-35.0f - Vn) * (1.0f / 9.0f));   // exp((-V-35)/9)
    float dN  = 1.0f - e1;
    float dM  = 1.0f - e2;
    float dNb = 1.0f - frcp(e1);                       // 1 - exp((V-25)/9)
    float dMb = 1.0f - frcp(e2);                       // 1 - exp((V+35)/9)
    float vm  = Vn - 25.0f, vp = Vn + 35.0f;
    float aN  = at25  ? 0.18f  :  0.02f  * vm * frcp(dN);
    float bN  = at25  ? 0.08f  : -0.002f * vm * frcp(dNb);
    float aM  = atm35 ? 1.638f :  0.182f * vp * frcp(dM);
    float bM  = atm35 ? 1.16f  : -0.124f * vp * frcp(dMb);
    float aH  = 0.25f * fexp((-90.0f - Vn) * (1.0f / 12.0f));
    float bH  = 0.25f * fexp((Vn + 34.0f)  * (1.0f / 12.0f));

    // ---- gate + synapse updates -----------------------------------------
    m = gate_update(aM, bM, m);
    n = gate_update(aN, bN, n);
    h = gate_update(aH, bH, h);
    float sy = 0.01f * (zprev + 0.1f);                 // A_D=1, A_R=0.1
    y = __builtin_fmaf(0.02f, zprev, (1.0f - sy) * y) * frcp(1.0f + sy);
    V = Vn;

    __builtin_nontemporal_store(vsig(V), &out[base + (size_t)t * (size_t)L_]);
  }
}

extern "C" void kernel_launch(void* const* d_in, const int* in_sizes, int n_in,
                              void* d_out, int out_size, void* d_ws, size_t ws_size,
                              hipStream_t stream) {
  (void)in_sizes; (void)n_in; (void)out_size; (void)d_ws; (void)ws_size;
  const float* z = (const float*)d_in[0];
  float* out = (float*)d_out;
  const int total = B_ * L_;                  // 16384 neurons, one thread each
  hh_synaptic_kernel<<<dim3(total / BLOCK), dim3(BLOCK), 0, stream>>>(z, out);
}